// GRAND_40802189312207
// MI455X (gfx1250) — compile-verified
//
#include <hip/hip_runtime.h>

#define N_NODES 50000
#define N_EDGES 1600000
#define N_GRAPHS 64
#define D_IN 128
#define D_H 512
#define D_OUTC 16

typedef __attribute__((ext_vector_type(2))) float v2f;
typedef __attribute__((ext_vector_type(8))) float v8f;

// workspace layout (in floats)
#define HA_OFF   0
#define HB_OFF   (N_NODES * D_IN)
#define DEG_OFF  (2 * N_NODES * D_IN)
#define CNT_OFF  (DEG_OFF + N_NODES)
#define POOL_OFF (CNT_OFF + 64)
#define TMP2_OFF (POOL_OFF + N_GRAPHS * D_H)

// ---- init: deg = 1 (self loop), cnt = 0, pool = 0 -------------------------
__global__ void grand_init(float* __restrict__ deg, float* __restrict__ cnt,
                           float* __restrict__ pool) {
    int i = blockIdx.x * blockDim.x + threadIdx.x;
    if (i < N_NODES) deg[i] = 1.0f;
    if (i < N_GRAPHS) cnt[i] = 0.0f;
    if (i < N_GRAPHS * D_H) pool[i] = 0.0f;
}

// ---- degree accumulation over edge rows -----------------------------------
__global__ void grand_deg(const long long* __restrict__ ei, float* __restrict__ deg) {
    int e = blockIdx.x * blockDim.x + threadIdx.x;   // grid exact: E/256
    int r = (int)ei[e];                              // edge_index[0][e]
    atomicAdd(&deg[r], 1.0f);
}

// ---- per-graph node counts ------------------------------------------------
__global__ void grand_cnt(const long long* __restrict__ batch, float* __restrict__ cnt) {
    int i = blockIdx.x * blockDim.x + threadIdx.x;
    if (i < N_NODES) atomicAdd(&cnt[(int)batch[i]], 1.0f);
}

// ---- deg -> 1/deg in place ------------------------------------------------
__global__ void grand_inv(float* __restrict__ deg) {
    int i = blockIdx.x * blockDim.x + threadIdx.x;
    if (i < N_NODES) deg[i] = 1.0f / deg[i];
}

// ---- self-loop term: out[i][:] = in[i][:] * invdeg[i] (also init of out) --
__global__ void grand_self(const float* __restrict__ in, const float* __restrict__ invd,
                           float* __restrict__ out) {
    int t = blockIdx.x * blockDim.x + threadIdx.x;   // N_NODES*32 threads
    int node = t >> 5;
    int j = (t & 31) << 2;
    float w = invd[node];
    float4 v = *(const float4*)(in + node * D_IN + j);
    float4 o;
    o.x = v.x * w; o.y = v.y * w; o.z = v.z * w; o.w = v.w * w;
    *(float4*)(out + node * D_IN + j) = o;
}

// ---- edge scatter: out[row] += in[col] * invdeg[row]; 1 wave per edge -----
__global__ void grand_edges(const long long* __restrict__ ei,
                            const float* __restrict__ invd,
                            const float* __restrict__ in,
                            float* __restrict__ out) {
    int gtid = blockIdx.x * blockDim.x + threadIdx.x;   // E*32 threads
    int e = gtid >> 5;                                   // wave-uniform
    int lane = threadIdx.x & 31;
    int r = (int)ei[e];
    int c = (int)ei[(long long)N_EDGES + e];
    float w = invd[r];
    float4 v = *(const float4*)(in + c * D_IN + (lane << 2));
    float* o = out + r * D_IN + (lane << 2);
    atomicAdd(o + 0, v.x * w);
    atomicAdd(o + 1, v.y * w);
    atomicAdd(o + 2, v.z * w);
    atomicAdd(o + 3, v.w * w);
}

// ---- GEMM1 (N x 128 @ 128 x 512) via fp32 WMMA, fused bias+relu+mean-pool -
__global__ void __launch_bounds__(256)
grand_gemm_pool(const float* __restrict__ h, const float* __restrict__ w1,
                const float* __restrict__ b1, const long long* __restrict__ batch,
                float* __restrict__ pool) {
    int wv   = threadIdx.x >> 5;
    int lane = threadIdx.x & 31;
    int tile = blockIdx.x * 8 + wv;      // (50000/16)*(512/16) = 100000 tiles
    int mt = tile >> 5;                  // / 32 N-tiles
    int nt = tile & 31;
    int half = lane >> 4;
    int ln = lane & 15;
    const int r0 = mt * 16;
    const int c0 = nt * 16;

    v8f acc = {0.f, 0.f, 0.f, 0.f, 0.f, 0.f, 0.f, 0.f};
    // A lane layout (16x4 f32): v0 = K={0|2}, v1 = K={1|3} -> contiguous f2
    const float* arow = h + (size_t)(r0 + ln) * D_IN + 2 * half;
    // B lane layout (4x16 f32): mirrored: rows K={2*half, 2*half+1}, col ln
    const float* bcol = w1 + (size_t)(2 * half) * D_H + c0 + ln;

    for (int k0 = 0; k0 < D_IN; k0 += 4) {
        v2f a = *(const v2f*)(arow + k0);
        v2f b;
        b.x = bcol[(size_t)k0 * D_H];
        b.y = bcol[(size_t)(k0 + 1) * D_H];
        acc = __builtin_amdgcn_wmma_f32_16x16x4_f32(
            /*neg_a=*/false, a, /*neg_b=*/false, b,
            /*c_mod=*/(short)0, acc, /*reuse_a=*/false, /*reuse_b=*/false);
    }

    float bias = b1[c0 + ln];
#pragma unroll
    for (int v = 0; v < 8; ++v) {
        int row = r0 + v + 8 * half;     // C/D layout: VGPR v = rows v, v+8
        float val = acc[v] + bias;
        val = val > 0.0f ? val : 0.0f;   // relu
        int g = (int)batch[row];
        atomicAdd(&pool[g * D_H + c0 + ln], val);
    }
}

// ---- tmp[g][j] = (sum_k pool[g][k] * w2[k][j]) / cnt[g] + b2[j] -----------
__global__ void grand_final1(const float* __restrict__ pool, const float* __restrict__ cnt,
                             const float* __restrict__ w2, const float* __restrict__ b2,
                             float* __restrict__ tmp) {
    int idx = blockIdx.x * blockDim.x + threadIdx.x;  // 64*512
    int g = idx >> 9;
    int j = idx & 511;
    const float* p = pool + (size_t)g * D_H;
    float s = 0.0f;
    for (int k = 0; k < D_H; ++k) s = fmaf(p[k], w2[(size_t)k * D_H + j], s);
    float c = cnt[g];
    c = c < 1.0f ? 1.0f : c;
    tmp[idx] = s / c + b2[j];
}

// ---- out[g][j] = sum_k tmp[g][k] * wc[k][j] + bc[j] -----------------------
__global__ void grand_final2(const float* __restrict__ tmp, const float* __restrict__ wc,
                             const float* __restrict__ bc, float* __restrict__ out) {
    int idx = blockIdx.x * blockDim.x + threadIdx.x;  // 64*16
    int g = idx >> 4;
    int j = idx & 15;
    const float* t = tmp + (size_t)g * D_H;
    float s = 0.0f;
    for (int k = 0; k < D_H; ++k) s = fmaf(t[k], wc[(size_t)k * D_OUTC + j], s);
    out[idx] = s + bc[j];
}

extern "C" void kernel_launch(void* const* d_in, const int* in_sizes, int n_in,
                              void* d_out, int out_size, void* d_ws, size_t ws_size,
                              hipStream_t stream) {
    const float*     x     = (const float*)d_in[0];
    const long long* ei    = (const long long*)d_in[1];  // int64 [2, E]
    const long long* batch = (const long long*)d_in[2];  // int64 [N]
    const float*     w1    = (const float*)d_in[3];
    const float*     b1    = (const float*)d_in[4];
    const float*     w2    = (const float*)d_in[5];
    const float*     b2    = (const float*)d_in[6];
    const float*     wc    = (const float*)d_in[7];
    const float*     bc    = (const float*)d_in[8];

    float* ws   = (float*)d_ws;
    float* hA   = ws + HA_OFF;
    float* hB   = ws + HB_OFF;
    float* deg  = ws + DEG_OFF;   // becomes 1/deg after grand_inv
    float* cnt  = ws + CNT_OFF;
    float* pool = ws + POOL_OFF;
    float* tmp2 = ws + TMP2_OFF;
    float* out  = (float*)d_out;

    const int nblk_nodes = (N_NODES + 255) / 256;

    grand_init<<<nblk_nodes, 256, 0, stream>>>(deg, cnt, pool);
    grand_deg<<<N_EDGES / 256, 256, 0, stream>>>(ei, deg);
    grand_cnt<<<nblk_nodes, 256, 0, stream>>>(batch, cnt);
    grand_inv<<<nblk_nodes, 256, 0, stream>>>(deg);

    // hop 1: x -> hA
    grand_self<<<N_NODES * 32 / 256, 256, 0, stream>>>(x, deg, hA);
    grand_edges<<<N_EDGES * 32 / 256, 256, 0, stream>>>(ei, deg, x, hA);
    // hop 2: hA -> hB
    grand_self<<<N_NODES * 32 / 256, 256, 0, stream>>>(hA, deg, hB);
    grand_edges<<<N_EDGES * 32 / 256, 256, 0, stream>>>(ei, deg, hA, hB);
    // hop 3: hB -> hA
    grand_self<<<N_NODES * 32 / 256, 256, 0, stream>>>(hB, deg, hA);
    grand_edges<<<N_EDGES * 32 / 256, 256, 0, stream>>>(ei, deg, hB, hA);

    // GEMM1 (WMMA fp32) + fused bias/relu/pool-sum
    grand_gemm_pool<<<(N_NODES / 16) * (D_H / 16) / 8, 256, 0, stream>>>(hA, w1, b1, batch, pool);

    // small tail: (pool/cnt) @ w2 + b2, then @ wc + bc
    grand_final1<<<N_GRAPHS * D_H / 256, 256, 0, stream>>>(pool, cnt, w2, b2, tmp2);
    grand_final2<<<N_GRAPHS * D_OUTC / 256, 256, 0, stream>>>(tmp2, wc, bc, out);
}